// SimpleTritonLayer_43104291782819
// MI455X (gfx1250) — compile-verified
//
#include <hip/hip_runtime.h>
#include <stdint.h>

// out[b,h] = fma(x[b,h], A[h], B[h]) where A/B fold lifecycle/strategy/blueprint.
// Memory-bound: 512 MB stream -> ~22us floor @ 23.3 TB/s. No matrix math; the
// CDNA5 feature that matters here is the async global->LDS copy engine
// (GLOBAL_LOAD_ASYNC_TO_LDS_B128 + ASYNCcnt) used as a deep prefetch pipeline.

typedef float v4f __attribute__((ext_vector_type(4)));

#define TPB   256          // 8 wave32 waves per block
#define TROWS 8            // rows staged per pipeline stage (ASYNCcnt <= 16)

// ---------------------------------------------------------------------------
// Phase 1: fold all per-column seed state into affine (A, B). 8192 threads.
// ---------------------------------------------------------------------------
__global__ void seedgate_build_ab(const int* __restrict__ lifecycle,
                                  const int* __restrict__ blueprint,
                                  const int* __restrict__ strategy,
                                  const float* __restrict__ weights,
                                  const int* __restrict__ chunk_ptr,
                                  float* __restrict__ A,
                                  float* __restrict__ B,
                                  int H) {
    int h = blockIdx.x * blockDim.x + threadIdx.x;
    if (h >= H) return;
    int chunk = chunk_ptr[0];
    int sid   = h / chunk;
    int life  = lifecycle[sid];
    float a = 1.0f, b = 0.0f;                 // inactive: out = x
    if (life >= 3 && life <= 6) {
        float wv = weights[(size_t)blueprint[sid] * (size_t)H + (size_t)h];
        int s = strategy[sid];
        if (s == 0)      { a = wv;                 }  // mul:   x*w
        else if (s == 1) { b = wv;                 }  // add:   x+w
        else             { a = 0.5f; b = 0.5f*wv;  }  // blend
    }
    A[h] = a;
    B[h] = b;
}

// ---------------------------------------------------------------------------
// Phase 2: streaming FMA with a double-buffered async global->LDS pipeline.
// Each lane owns one float4 column; each block streams rows_per_block rows.
// ---------------------------------------------------------------------------
__device__ __forceinline__ void async_copy_b128(uint32_t lds_addr, uint64_t gaddr) {
    // GLOBAL_LOAD_ASYNC_TO_LDS_B128, GV addressing (vaddr = 64-bit VGPR pair).
    // VDST VGPR carries the per-lane LDS byte address. Tracked by ASYNCcnt.
    asm volatile("global_load_async_to_lds_b128 %0, %1, off"
                 :: "v"(lds_addr), "v"(gaddr)
                 : "memory");
}

__global__ void seedgate_apply(const v4f* __restrict__ x,
                               v4f* __restrict__ out,
                               const v4f* __restrict__ A4,
                               const v4f* __restrict__ B4,
                               int cols4, int rows_per_block) {
    __shared__ v4f sbuf[2][TROWS][TPB];       // 64 KB of the WGP's 320 KB LDS

    const int tid  = threadIdx.x;
    const int col4 = blockIdx.x * TPB + tid;  // < cols4 by launch construction

    v4f a = A4[col4];                         // temporal: stays hot in L2/WGP$
    v4f b = B4[col4];

    const size_t stride = (size_t)cols4;
    size_t idx = (size_t)blockIdx.y * (size_t)rows_per_block * stride + (size_t)col4;

    // Flat addresses into the LDS aperture carry the LDS byte offset in [31:0].
    const uint32_t lbase    = (uint32_t)(uintptr_t)(&sbuf[0][0][tid]);
    const uint32_t lrow     = TPB * (uint32_t)sizeof(v4f);   // bytes per staged row
    const uint32_t bufbytes = TROWS * lrow;                  // bytes per buffer

    // Prologue: enqueue stage 0 (TROWS async B128 copies per lane's column).
    #pragma unroll
    for (int r = 0; r < TROWS; ++r) {
        async_copy_b128(lbase + (uint32_t)r * lrow,
                        (uint64_t)(uintptr_t)(x + idx + (size_t)r * stride));
    }

    for (int base = 0; base < rows_per_block; base += TROWS) {
        const int cur = (base / TROWS) & 1;

        if (base + TROWS < rows_per_block) {
            // Enqueue next stage into the other buffer, then drain this stage:
            // after issuing TROWS more, ASYNCcnt <= TROWS means stage `cur` landed.
            const int nxt = cur ^ 1;
            #pragma unroll
            for (int r = 0; r < TROWS; ++r) {
                async_copy_b128(lbase + (uint32_t)nxt * bufbytes + (uint32_t)r * lrow,
                                (uint64_t)(uintptr_t)(x + idx
                                    + (size_t)(base + TROWS + r) * stride));
            }
            asm volatile("s_wait_asynccnt 8" ::: "memory");   // == TROWS
        } else {
            asm volatile("s_wait_asynccnt 0" ::: "memory");   // final stage
        }

        // Consume stage `cur`: ds_load_b128 -> 4x FMA -> NT b128 store.
        #pragma unroll
        for (int r = 0; r < TROWS; ++r) {
            v4f xv = sbuf[cur][r][tid];
            v4f y;
            y.x = fmaf(xv.x, a.x, b.x);
            y.y = fmaf(xv.y, a.y, b.y);
            y.z = fmaf(xv.z, a.z, b.z);
            y.w = fmaf(xv.w, a.w, b.w);
            __builtin_nontemporal_store(y, out + idx + (size_t)(base + r) * stride);
        }
    }
}

// ---------------------------------------------------------------------------
extern "C" void kernel_launch(void* const* d_in, const int* in_sizes, int n_in,
                              void* d_out, int out_size, void* d_ws, size_t ws_size,
                              hipStream_t stream) {
    // Input order per setup_inputs(): x, lifecycle, blueprint, strategy, weights, chunk_size
    const float* x         = (const float*)d_in[0];
    const int*   lifecycle = (const int*)  d_in[1];
    const int*   blueprint = (const int*)  d_in[2];
    const int*   strategy  = (const int*)  d_in[3];
    const float* weights   = (const float*)d_in[4];
    const int*   chunk     = (const int*)  d_in[5];
    float*       out       = (float*)d_out;

    const int H    = 8192;                                   // HIDDEN
    const int rows = (int)((long long)in_sizes[0] / H);      // 8192

    // Workspace: A[H] then B[H] (64 KB), rebuilt every call (deterministic).
    float* A = (float*)d_ws;
    float* B = A + H;

    seedgate_build_ab<<<(H + TPB - 1) / TPB, TPB, 0, stream>>>(
        lifecycle, blueprint, strategy, weights, chunk, A, B, H);

    const int cols4 = H / 4;                 // 2048 float4 columns
    const int gx    = cols4 / TPB;           // 8 blocks across a row
    int rows_per_block = 32;                 // 4 pipeline stages of TROWS=8
    while (rows_per_block > TROWS && (rows % rows_per_block)) rows_per_block >>= 1;
    const int gy = rows / rows_per_block;    // 256  -> 2048 blocks total

    seedgate_apply<<<dim3(gx, gy), TPB, 0, stream>>>(
        (const v4f*)x, (v4f*)out, (const v4f*)A, (const v4f*)B,
        cols4, rows_per_block);
}